// GumbelQuantizer_48206712930873
// MI455X (gfx1250) — compile-verified
//
#include <hip/hip_runtime.h>
#include <stdint.h>

// Shapes from reference: z,g = (8, 8192, 32, 32) f32; W = (8192, 256) f32
// Outputs concatenated: z_q (8*256*1024) | diff (1) | ind (8*1024)
#define BATCH   8
#define NDIM    8192
#define HW      1024
#define ZQ_SIZE (BATCH * 256 * HW)    // 2097152
#define DIFF_OFF ZQ_SIZE
#define IND_OFF (ZQ_SIZE + 1)
#define LOG_N   9.010913347279289f    // ln(8192)
#define KL_SCALE (5e-4f / 8192.0f)    // KL_WEIGHT / (b*h*w)

// ---- CDNA5 async global->LDS copy (each lane moves 16B, non-temporal) -----
__device__ __forceinline__ void async_ld16(uint32_t lds_off, const float* gsrc) {
  asm volatile("global_load_async_to_lds_b128 %0, %1, off th:TH_LOAD_NT"
               :: "v"(lds_off), "v"((uint64_t)(uintptr_t)gsrc)
               : "memory");
}
__device__ __forceinline__ void wait_async_le2() {
  asm volatile("s_wait_asynccnt 2" ::: "memory");
}
__device__ __forceinline__ void wait_async_0() {
  asm volatile("s_wait_asynccnt 0" ::: "memory");
}
// Guard: prior ds_loads of the buffer being refilled must have completed
// before the async engine may overwrite it (async writes are unordered
// w.r.t. in-flight LDS reads).
__device__ __forceinline__ void wait_ds_0() {
  asm volatile("s_wait_dscnt 0" ::: "memory");
}

// ---- Kernel 1: streaming per-(batch, n-chunk) partial reduction -----------
// block = (b, c); 256 threads x 4 positions; loops over chn n-values.
// Double-buffered LDS tiles filled with CDNA5 async global->LDS loads; each
// lane consumes exactly the 16B it loaded, so only per-wave ASYNCcnt is needed.
// cshift = log2(nchunk); nchunk/chn/plane all derived via shifts (no SALU div).
__global__ void __launch_bounds__(256)
gq_partial(const float* __restrict__ z, const float* __restrict__ g,
           float* __restrict__ ws, int cshift) {
  __shared__ float zb[2][HW];
  __shared__ float gb[2][HW];
  const int t = threadIdx.x;
  const int b = blockIdx.x >> cshift;
  const int c = blockIdx.x & ((1 << cshift) - 1);
  const int chn = NDIM >> cshift;
  const size_t base = ((size_t)blockIdx.x * chn) * HW + 4u * t; // == (b*NDIM + c*chn)*HW
  const float* zs = z + base;
  const float* gs = g + base;

  // LDS byte offsets for this thread's 16B slots (low 32 bits of generic addr)
  const uint32_t zl0 = (uint32_t)(uintptr_t)&zb[0][4 * t];
  const uint32_t zl1 = (uint32_t)(uintptr_t)&zb[1][4 * t];
  const uint32_t gl0 = (uint32_t)(uintptr_t)&gb[0][4 * t];
  const uint32_t gl1 = (uint32_t)(uintptr_t)&gb[1][4 * t];

  async_ld16(zl0, zs);
  async_ld16(gl0, gs);

  float maxv[4], S[4], U[4];
  int maxi[4];
#pragma unroll
  for (int j = 0; j < 4; ++j) { maxv[j] = -3.0e38f; S[j] = 0.f; U[j] = 0.f; maxi[j] = 0; }

  const int nbase = c * chn;
  for (int k = 0; k < chn; ++k) {
    const int s = k & 1;
    if (k + 1 < chn) {
      wait_ds_0();                       // WAR: finish reads of buffer s^1
      async_ld16(s ? zl0 : zl1, zs + (size_t)(k + 1) * HW);
      async_ld16(s ? gl0 : gl1, gs + (size_t)(k + 1) * HW);
      wait_async_le2();                  // the 2 ops for buffer `s` are done
    } else {
      wait_async_0();
    }
    const float4 zv = *(const float4*)&zb[s][4 * t];
    const float4 gv = *(const float4*)&gb[s][4 * t];
    const int n = nbase + k;
    const float za[4] = {zv.x, zv.y, zv.z, zv.w};
    const float ga[4] = {gv.x, gv.y, gv.z, gv.w};
#pragma unroll
    for (int j = 0; j < 4; ++j) {
      const float l = za[j] + ga[j];     // logits (TEMP == 1)
      if (l > maxv[j]) { maxv[j] = l; maxi[j] = n; }  // first-max == argmax
      const float e = __expf(za[j]);     // N(0,1) inputs: no overflow
      S[j] += e;
      U[j] = fmaf(za[j], e, U[j]);
    }
  }

  const size_t plane = (size_t)(BATCH << cshift) * HW;
  const size_t pc = ((size_t)blockIdx.x) * HW + 4u * t;   // == (b*nchunk + c)*HW + 4t
  *(float4*)&ws[0 * plane + pc] = make_float4(maxv[0], maxv[1], maxv[2], maxv[3]);
  *(float4*)&ws[1 * plane + pc] = make_float4((float)maxi[0], (float)maxi[1],
                                              (float)maxi[2], (float)maxi[3]);
  *(float4*)&ws[2 * plane + pc] = make_float4(S[0], S[1], S[2], S[3]);
  *(float4*)&ws[3 * plane + pc] = make_float4(U[0], U[1], U[2], U[3]);
}

// ---- Kernel 2: merge chunks per position; emit ind + per-block KL sums ----
__global__ void __launch_bounds__(256)
gq_reduce(const float* __restrict__ ws, float* __restrict__ out,
          float* __restrict__ blocksums, int cshift) {
  const int tid = blockIdx.x * 256 + threadIdx.x;  // 0..8191 = (b,pos)
  const int b = tid >> 10, pos = tid & 1023;
  const int nchunk = 1 << cshift;
  const size_t plane = (size_t)(BATCH << cshift) * HW;
  const float* P0 = ws;
  const float* P1 = ws + plane;
  const float* P2 = ws + 2 * plane;
  const float* P3 = ws + 3 * plane;
  float mv = -3.0e38f, mi = 0.f, S = 0.f, U = 0.f;
  for (int c = 0; c < nchunk; ++c) {
    const size_t idx = (size_t)((b << cshift) + c) * HW + pos;
    const float v = P0[idx];
    const float ii = P1[idx];
    if (v > mv) { mv = v; mi = ii; }     // strict > keeps earliest n on ties
    S += P2[idx];
    U += P3[idx];
  }
  out[IND_OFF + tid] = mi;               // argmax index (exact integer in f32)
  const float kl = LOG_N + U / S - __logf(S);

  __shared__ float red[256];
  red[threadIdx.x] = kl;
  __syncthreads();
  for (int s = 128; s > 0; s >>= 1) {
    if (threadIdx.x < s) red[threadIdx.x] += red[threadIdx.x + s];
    __syncthreads();
  }
  if (threadIdx.x == 0) blocksums[blockIdx.x] = red[0];
}

// ---- Kernel 3: finalize diff scalar (deterministic serial sum of 32) ------
__global__ void gq_final(const float* __restrict__ blocksums, float* __restrict__ out) {
  if (threadIdx.x == 0 && blockIdx.x == 0) {
    float s = 0.f;
    for (int i = 0; i < 32; ++i) s += blocksums[i];
    out[DIFF_OFF] = KL_SCALE * s;
  }
}

// ---- Kernel 4: z_q gather: z_q[b,d,pos] = W[ind[b,pos], d] ----------------
// block = (b, d); threads sweep pos -> coalesced float4 stores; W gathers hit L2.
__global__ void __launch_bounds__(256)
gq_gather(const float* __restrict__ W, const float* __restrict__ indf,
          float* __restrict__ zq) {
  const int b = blockIdx.x >> 8;
  const int d = blockIdx.x & 255;
  const int t = threadIdx.x;
  const float4 iv = *(const float4*)&indf[b * HW + 4 * t];
  float4 o;
  o.x = W[(int)iv.x * 256 + d];
  o.y = W[(int)iv.y * 256 + d];
  o.z = W[(int)iv.z * 256 + d];
  o.w = W[(int)iv.w * 256 + d];
  *(float4*)&zq[(size_t)b * 256 * HW + (size_t)d * HW + 4 * t] = o;
}

extern "C" void kernel_launch(void* const* d_in, const int* in_sizes, int n_in,
                              void* d_out, int out_size, void* d_ws, size_t ws_size,
                              hipStream_t stream) {
  (void)in_sizes; (void)n_in; (void)out_size;
  const float* z  = (const float*)d_in[0];  // (8, 8192, 32, 32)
  const float* W  = (const float*)d_in[1];  // (8192, 256)
  const float* gn = (const float*)d_in[2];  // (8, 8192, 32, 32)
  float* out = (float*)d_out;
  float* ws  = (float*)d_ws;

  // Pick the largest power-of-two chunking (=> most blocks) fitting in d_ws.
  int cshift = 3;                                    // nchunk = 8 floor
  for (int cs = 7; cs >= 3; --cs) {                  // try nchunk = 128..8
    const size_t need = ((size_t)4 * (BATCH << cs) * HW + 32) * sizeof(float);
    if (ws_size >= need) { cshift = cs; break; }
  }
  const int nchunk = 1 << cshift;
  float* blocksums = ws + (size_t)4 * (BATCH << cshift) * HW;

  gq_partial<<<BATCH * nchunk, 256, 0, stream>>>(z, gn, ws, cshift);
  gq_reduce<<<(BATCH * HW) / 256, 256, 0, stream>>>(ws, out, blocksums, cshift);
  gq_final<<<1, 32, 0, stream>>>(blocksums, out);
  gq_gather<<<BATCH * 256, 256, 0, stream>>>(W, out + IND_OFF, out);
}